// DBISSF_Attention_11330123727073
// MI455X (gfx1250) — compile-verified
//
#include <hip/hip_runtime.h>
#include <math.h>
#include <stdint.h>

#define BATCH  4
#define SEQ    16384
#define DIN    192
#define NST    4
#define RK     6
#define PJ     14
#define CHUNK  256
#define NCHUNK 64      // SEQ / CHUNK
#define XPAD   196     // LDS row stride (== 4 mod 64 -> conflict-free)

typedef __attribute__((ext_vector_type(2))) float v2f;
typedef __attribute__((ext_vector_type(8))) float v8f;

__device__ __forceinline__ v8f wmma_f32_16x16x4(v2f a, v2f b, v8f c) {
  return __builtin_amdgcn_wmma_f32_16x16x4_f32(false, a, false, b, (short)0, c,
                                               false, false);
}

__device__ __forceinline__ float softplus_f(float x) {
  return (x > 20.0f) ? x : log1pf(__expf(x));
}

// Async global->LDS staging of one 16-byte packet (tracked by ASYNCcnt).
__device__ __forceinline__ void async_b128_to_lds(const float* gptr,
                                                  float* lptr) {
  uint64_t ga = (uint64_t)(uintptr_t)gptr;
  uint32_t la = (uint32_t)(uintptr_t)lptr;  // addrspace(3) offset in low bits
  asm volatile("global_load_async_to_lds_b128 %0, %1, off"
               :: "v"(la), "v"(ga) : "memory");
}
__device__ __forceinline__ void wait_async0() {
  asm volatile("s_wait_asynccnt 0x0" ::: "memory");
}

// -------------------------------------------------------------------------
// Pass 1: projections via V_WMMA_F32_16X16X4_F32, everything LDS-staged.
//   dbl[16x14] = X[16x192] @ W^T        (K=192 -> 48 WMMAs)
//   dt [16x192]= dbl[:, :6] @ dtW^T     (K padded 6->8 -> 2 WMMAs x 12 tiles)
// One wave per 16-token tile; 4 waves/block all share one stream.
// -------------------------------------------------------------------------
__global__ __launch_bounds__(128) void proj_wmma_kernel(
    const float* __restrict__ x_rgb, const float* __restrict__ x_srgb,
    const float* __restrict__ x_e,   const float* __restrict__ x_se,
    const float* __restrict__ xp1, const float* __restrict__ xp2,
    const float* __restrict__ xps,
    const float* __restrict__ dt1w, const float* __restrict__ dt2w,
    const float* __restrict__ dtsw,
    float* __restrict__ dt_ws, float* __restrict__ B_ws,
    float* __restrict__ C_ws)
{
  __shared__ float Xs[4][16][XPAD];   // per-wave X tile (padded rows)
  __shared__ float Wp[16][XPAD];      // xproj weights, rows >= 14 zeroed
  __shared__ float DWt[8][DIN];       // dt-proj weights transposed, k >= 6 zero
  __shared__ float dbl[4][16][16];    // GEMM1 result tile per wave
  __shared__ float dtA[4][16][8];     // dbl[:, :6] zero-padded to K=8

  const int wv   = threadIdx.x >> 5;
  const int lane = threadIdx.x & 31;
  const int row  = lane & 15;
  const int hi   = lane >> 4;

  const int g    = blockIdx.x * 4 + wv;     // 0 .. 3*4096-1
  const int st   = g / 4096;                // block-uniform
  const int tile = g - st * 4096;
  const int b    = tile >> 10;
  const int l0   = (tile & 1023) << 4;

  const float* X  = (st == 0) ? x_rgb : (st == 1) ? x_e : x_srgb;
  const float* W  = (st == 0) ? xp1 : (st == 1) ? xp2 : xps;
  const float* DW = (st == 0) ? dt1w : (st == 1) ? dt2w : dtsw;

  // ---- stage weights (block-wide, zero-padded) ----
  for (int i = threadIdx.x; i < 16 * XPAD; i += 128) {
    const int r = i / XPAD, cc = i - r * XPAD;
    Wp[r][cc] = (r < PJ && cc < DIN) ? W[r * DIN + cc] : 0.0f;
  }
  for (int i = threadIdx.x; i < 8 * DIN; i += 128) {
    const int k = i / DIN, n = i - k * DIN;
    DWt[k][n] = (k < RK) ? DW[n * RK + k] : 0.0f;
  }

  // ---- stage X tile per wave with coalesced float4 loads ----
  const float* Xbase = X + (size_t)(b * SEQ + l0) * DIN;
  if (st == 2) {
    const float* X2base = x_se + (size_t)(b * SEQ + l0) * DIN;
    for (int i = lane; i < 16 * 48; i += 32) {
      const int r = i / 48, c4 = i - r * 48;
      float4 xv = ((const float4*)(Xbase + r * DIN))[c4];
      float4 x2 = ((const float4*)(X2base + r * DIN))[c4];
      xv.x += x2.x; xv.y += x2.y; xv.z += x2.z; xv.w += x2.w;
      ((float4*)&Xs[wv][r][0])[c4] = xv;
    }
  } else {
    for (int i = lane; i < 16 * 48; i += 32) {
      const int r = i / 48, c4 = i - r * 48;
      ((float4*)&Xs[wv][r][0])[c4] = ((const float4*)(Xbase + r * DIN))[c4];
    }
  }
  __syncthreads();

  // ---- GEMM1: dbl = X * W^T (all operand fetches unconditional LDS b64) ----
  v8f acc = {};
  for (int k = 0; k < DIN; k += 4) {
    const int kk = k + hi * 2;
    const v2f af = *(const v2f*)&Xs[wv][row][kk];
    const v2f bf = *(const v2f*)&Wp[row][kk];
    acc = wmma_f32_16x16x4(af, bf, acc);
  }
  // C layout: VGPR j -> (M = j + hi*8, N = row)
#pragma unroll
  for (int j = 0; j < 8; ++j) {
    const float val = acc[j];
    dbl[wv][hi * 8 + j][row] = val;
    if (row < 8) dtA[wv][hi * 8 + j][row] = (row < RK) ? val : 0.0f;
  }
  __syncthreads();

  // ---- write B, C (dbl cols 6..9 / 10..13) as float4 per token ----
  if (lane < 16) {
    const int m = lane;
    const size_t tok = (size_t)(st * BATCH + b) * SEQ + (l0 + m);
    float4 Bv = make_float4(dbl[wv][m][RK + 0], dbl[wv][m][RK + 1],
                            dbl[wv][m][RK + 2], dbl[wv][m][RK + 3]);
    float4 Cv = make_float4(dbl[wv][m][RK + 4], dbl[wv][m][RK + 5],
                            dbl[wv][m][RK + 6], dbl[wv][m][RK + 7]);
    reinterpret_cast<float4*>(B_ws)[tok] = Bv;
    reinterpret_cast<float4*>(C_ws)[tok] = Cv;
  }

  // ---- GEMM2: dt = dbl[:, :6] * DW^T (K padded to 8, zero-padded LDS) ----
  const size_t sb = (size_t)(st * BATCH + b) * SEQ;
  for (int nt = 0; nt < 12; ++nt) {
    v8f acc2 = {};
#pragma unroll
    for (int ks = 0; ks < 2; ++ks) {
      const int kk = ks * 4 + hi * 2;
      const v2f af = *(const v2f*)&dtA[wv][row][kk];
      const int n = nt * 16 + row;
      const v2f bf = {DWt[kk][n], DWt[kk + 1][n]};
      acc2 = wmma_f32_16x16x4(af, bf, acc2);
    }
#pragma unroll
    for (int j = 0; j < 8; ++j) {
      const int tok = l0 + hi * 8 + j;
      dt_ws[(sb + tok) * DIN + nt * 16 + row] = acc2[j];
    }
  }
}

// -------------------------------------------------------------------------
// Scans s=0..3 (block = one (s, b, chunk); 192 threads = all channels):
//   s0: u=x_rgb, dt/B stream0, C stream1, A1/D1/b1  -> out slot 0
//   s1: u=x_e,   dt/B stream1, C stream0, A2/D2/b2  -> out slot 2
//   s2: u=x_rgb, dt/B/C stream2, As/Ds/bs           -> out slot 1
//   s3: u=x_e,   dt/B/C stream2, As/Ds/bs           -> out slot 3
// -------------------------------------------------------------------------

// Pass 2: per-chunk carries (prod a, chunk-local state). B staged via async.
__global__ __launch_bounds__(192) void scan_phase1_kernel(
    const float* __restrict__ x_rgb, const float* __restrict__ x_e,
    const float* __restrict__ dt_ws, const float* __restrict__ B_ws,
    const float* __restrict__ A_log1, const float* __restrict__ A_log2,
    const float* __restrict__ A_logs,
    const float* __restrict__ dt1b, const float* __restrict__ dt2b,
    const float* __restrict__ dtsb,
    float* __restrict__ carryA, float* __restrict__ carryS)
{
  __shared__ float shB[CHUNK * 4];
  const int d = threadIdx.x;
  const int c = blockIdx.x % NCHUNK;
  const int b = (blockIdx.x / NCHUNK) % BATCH;
  const int s = blockIdx.x / (NCHUNK * BATCH);
  const int l0 = c * CHUNK;

  const int stp = (s <= 1) ? s : 2;
  const float* U   = (s == 1 || s == 3) ? x_e : x_rgb;
  const float* AL  = (s == 0) ? A_log1 : (s == 1) ? A_log2 : A_logs;
  const float* DTB = (s == 0) ? dt1b : (s == 1) ? dt2b : dtsb;

  const size_t sb = (size_t)(stp * BATCH + b) * SEQ;
  const float* Bc = B_ws + sb * 4 + (size_t)l0 * 4;

  // async-stage this chunk's B stream (256 tokens * 16B) into LDS
  async_b128_to_lds(Bc + d * 4, &shB[d * 4]);
  if (d < CHUNK - 192) async_b128_to_lds(Bc + (192 + d) * 4, &shB[(192 + d) * 4]);
  wait_async0();
  __syncthreads();

  float An[NST];
#pragma unroll
  for (int n = 0; n < NST; ++n) An[n] = -__expf(AL[d * NST + n]);
  const float bias = DTB[d];

  const float* dtp = dt_ws + sb * DIN + d;
  const float* up  = U + (size_t)b * SEQ * DIN + d;

  float ap[NST] = {1.f, 1.f, 1.f, 1.f};
  float S[NST]  = {0.f, 0.f, 0.f, 0.f};
  for (int l = l0; l < l0 + CHUNK; ++l) {
    if (l + 8 < SEQ) {
      __builtin_prefetch(dtp + (size_t)(l + 8) * DIN, 0, 1);
      __builtin_prefetch(up + (size_t)(l + 8) * DIN, 0, 1);
    }
    const float delta = softplus_f(dtp[(size_t)l * DIN] + bias);
    const float u = up[(size_t)l * DIN];
    const float4 Bv = ((const float4*)shB)[l - l0];
    const float bvals[NST] = {Bv.x, Bv.y, Bv.z, Bv.w};
#pragma unroll
    for (int n = 0; n < NST; ++n) {
      const float a = __expf(delta * An[n]);
      S[n] = a * S[n] + delta * bvals[n] * u;
      ap[n] *= a;
    }
  }
  const size_t idx = ((((size_t)(s * BATCH + b) * DIN + d) * NCHUNK) + c) * NST;
#pragma unroll
  for (int n = 0; n < NST; ++n) { carryA[idx + n] = ap[n]; carryS[idx + n] = S[n]; }
}

// Pass 3: sequential combine over 64 chunks -> incoming state per chunk.
__global__ __launch_bounds__(256) void scan_combine_kernel(
    const float* __restrict__ carryA, const float* __restrict__ carryS,
    float* __restrict__ hin)
{
  const int tid = blockIdx.x * 256 + threadIdx.x;  // 0..12287 = (s,b,d,n)
  const int n = tid & 3;
  const size_t base = (size_t)(tid >> 2) * NCHUNK * NST + n;
  float h = 0.f;
  for (int c = 0; c < NCHUNK; ++c) {
    const size_t idx = base + (size_t)c * NST;
    hin[idx] = h;
    h = carryA[idx] * h + carryS[idx];
  }
}

// Pass 4: replay chunks with known incoming state. B and C staged via async.
__global__ __launch_bounds__(192) void scan_phase2_kernel(
    const float* __restrict__ x_rgb, const float* __restrict__ x_e,
    const float* __restrict__ dt_ws, const float* __restrict__ B_ws,
    const float* __restrict__ C_ws,
    const float* __restrict__ A_log1, const float* __restrict__ A_log2,
    const float* __restrict__ A_logs,
    const float* __restrict__ dt1b, const float* __restrict__ dt2b,
    const float* __restrict__ dtsb,
    const float* __restrict__ D1, const float* __restrict__ D2,
    const float* __restrict__ Ds,
    const float* __restrict__ hin, float* __restrict__ out)
{
  __shared__ float shB[CHUNK * 4];
  __shared__ float shC[CHUNK * 4];
  const int d = threadIdx.x;
  const int c = blockIdx.x % NCHUNK;
  const int b = (blockIdx.x / NCHUNK) % BATCH;
  const int s = blockIdx.x / (NCHUNK * BATCH);
  const int l0 = c * CHUNK;

  const int stp  = (s <= 1) ? s : 2;
  const int cst  = (s == 0) ? 1 : (s == 1) ? 0 : 2;
  const int slot = (s == 0) ? 0 : (s == 1) ? 2 : (s == 2) ? 1 : 3;
  const float* U   = (s == 1 || s == 3) ? x_e : x_rgb;
  const float* AL  = (s == 0) ? A_log1 : (s == 1) ? A_log2 : A_logs;
  const float* DTB = (s == 0) ? dt1b : (s == 1) ? dt2b : dtsb;
  const float* Dp  = (s == 0) ? D1 : (s == 1) ? D2 : Ds;

  const size_t sb = (size_t)(stp * BATCH + b) * SEQ;
  const float* Bc = B_ws + sb * 4 + (size_t)l0 * 4;
  const float* Cc = C_ws + (size_t)(cst * BATCH + b) * SEQ * 4 + (size_t)l0 * 4;

  async_b128_to_lds(Bc + d * 4, &shB[d * 4]);
  async_b128_to_lds(Cc + d * 4, &shC[d * 4]);
  if (d < CHUNK - 192) {
    async_b128_to_lds(Bc + (192 + d) * 4, &shB[(192 + d) * 4]);
    async_b128_to_lds(Cc + (192 + d) * 4, &shC[(192 + d) * 4]);
  }
  wait_async0();
  __syncthreads();

  float An[NST];
#pragma unroll
  for (int n = 0; n < NST; ++n) An[n] = -__expf(AL[d * NST + n]);
  const float bias = DTB[d];
  const float Dd = Dp[d];

  const float* dtp = dt_ws + sb * DIN + d;
  const float* up  = U + (size_t)b * SEQ * DIN + d;
  float* op = out + (size_t)slot * BATCH * SEQ * DIN + (size_t)b * SEQ * DIN + d;

  float h[NST];
  const size_t hidx = ((((size_t)(s * BATCH + b) * DIN + d) * NCHUNK) + c) * NST;
#pragma unroll
  for (int n = 0; n < NST; ++n) h[n] = hin[hidx + n];

  for (int l = l0; l < l0 + CHUNK; ++l) {
    if (l + 8 < SEQ) {
      __builtin_prefetch(dtp + (size_t)(l + 8) * DIN, 0, 1);
      __builtin_prefetch(up + (size_t)(l + 8) * DIN, 0, 1);
    }
    const float delta = softplus_f(dtp[(size_t)l * DIN] + bias);
    const float u = up[(size_t)l * DIN];
    const float4 Bv = ((const float4*)shB)[l - l0];
    const float4 Cv = ((const float4*)shC)[l - l0];
    const float bvals[NST] = {Bv.x, Bv.y, Bv.z, Bv.w};
    const float cvals[NST] = {Cv.x, Cv.y, Cv.z, Cv.w};
    float y = Dd * u;
#pragma unroll
    for (int n = 0; n < NST; ++n) {
      const float a = __expf(delta * An[n]);
      h[n] = a * h[n] + delta * bvals[n] * u;
      y += h[n] * cvals[n];
    }
    op[(size_t)l * DIN] = y;
  }
}

// Pass 5: in-place LayerNorm over D_INNER=192; one wave32 per (slot,b,l) row.
__global__ __launch_bounds__(256) void ln_kernel(
    float* __restrict__ out,
    const float* __restrict__ ln1w, const float* __restrict__ ln1b,
    const float* __restrict__ ln2w, const float* __restrict__ ln2b,
    const float* __restrict__ lnsw, const float* __restrict__ lnsb)
{
  const int row = blockIdx.x * 8 + (threadIdx.x >> 5);
  const int lane = threadIdx.x & 31;
  const int slot = row / (BATCH * SEQ);
  float* rp = out + (size_t)row * DIN;

  float v[6];
  float sum = 0.f;
#pragma unroll
  for (int k = 0; k < 6; ++k) { v[k] = rp[k * 32 + lane]; sum += v[k]; }
#pragma unroll
  for (int off = 16; off > 0; off >>= 1) sum += __shfl_xor(sum, off, 32);
  const float mu = sum * (1.0f / DIN);

  float s2 = 0.f;
#pragma unroll
  for (int k = 0; k < 6; ++k) { const float t = v[k] - mu; s2 += t * t; }
#pragma unroll
  for (int off = 16; off > 0; off >>= 1) s2 += __shfl_xor(s2, off, 32);
  const float inv = rsqrtf(s2 * (1.0f / DIN) + 1e-5f);

  const float* w  = (slot == 0) ? ln1w : (slot == 2) ? ln2w : lnsw;
  const float* bb = (slot == 0) ? ln1b : (slot == 2) ? ln2b : lnsb;
#pragma unroll
  for (int k = 0; k < 6; ++k) {
    const int dd = k * 32 + lane;
    rp[dd] = (v[k] - mu) * inv * w[dd] + bb[dd];
  }
}

// -------------------------------------------------------------------------
extern "C" void kernel_launch(void* const* d_in, const int* in_sizes, int n_in,
                              void* d_out, int out_size, void* d_ws,
                              size_t ws_size, hipStream_t stream)
{
  const float* x_rgb  = (const float*)d_in[0];
  const float* x_srgb = (const float*)d_in[1];
  const float* x_e    = (const float*)d_in[2];
  const float* x_se   = (const float*)d_in[3];
  const float* xp1    = (const float*)d_in[4];
  const float* xp2    = (const float*)d_in[5];
  const float* xps    = (const float*)d_in[6];
  const float* dt1w   = (const float*)d_in[7];
  const float* dt2w   = (const float*)d_in[8];
  const float* dtsw   = (const float*)d_in[9];
  const float* dt1b   = (const float*)d_in[10];
  const float* dt2b   = (const float*)d_in[11];
  const float* dtsb   = (const float*)d_in[12];
  const float* A1     = (const float*)d_in[13];
  const float* A2     = (const float*)d_in[14];
  const float* As     = (const float*)d_in[15];
  const float* D1     = (const float*)d_in[16];
  const float* D2     = (const float*)d_in[17];
  const float* Ds     = (const float*)d_in[18];
  const float* ln1w   = (const float*)d_in[19];
  const float* ln1b   = (const float*)d_in[20];
  const float* ln2w   = (const float*)d_in[21];
  const float* ln2b   = (const float*)d_in[22];
  const float* lnsw   = (const float*)d_in[23];
  const float* lnsb   = (const float*)d_in[24];
  float* out = (float*)d_out;

  float* ws = (float*)d_ws;
  float* dt_ws  = ws;                                          // 3*B*SEQ*DIN
  float* B_ws   = dt_ws + (size_t)3 * BATCH * SEQ * DIN;       // 3*B*SEQ*4
  float* C_ws   = B_ws + (size_t)3 * BATCH * SEQ * 4;          // 3*B*SEQ*4
  float* carryA = C_ws + (size_t)3 * BATCH * SEQ * 4;          // 4*B*DIN*NCHUNK*4
  float* carryS = carryA + (size_t)4 * BATCH * DIN * NCHUNK * NST;
  float* hin    = carryS + (size_t)4 * BATCH * DIN * NCHUNK * NST;

  proj_wmma_kernel<<<3072, 128, 0, stream>>>(
      x_rgb, x_srgb, x_e, x_se, xp1, xp2, xps, dt1w, dt2w, dtsw,
      dt_ws, B_ws, C_ws);

  scan_phase1_kernel<<<4 * BATCH * NCHUNK, 192, 0, stream>>>(
      x_rgb, x_e, dt_ws, B_ws, A1, A2, As, dt1b, dt2b, dtsb, carryA, carryS);

  scan_combine_kernel<<<48, 256, 0, stream>>>(carryA, carryS, hin);

  scan_phase2_kernel<<<4 * BATCH * NCHUNK, 192, 0, stream>>>(
      x_rgb, x_e, dt_ws, B_ws, C_ws, A1, A2, As, dt1b, dt2b, dtsb,
      D1, D2, Ds, hin, out);

  ln_kernel<<<32768, 256, 0, stream>>>(out, ln1w, ln1b, ln2w, ln2b, lnsw, lnsb);
}